// BackProjection_12438225290006
// MI455X (gfx1250) — compile-verified
//
#include <hip/hip_runtime.h>

// PET-style Gaussian backprojection on MI455X (gfx1250):
//  - one workgroup per output slice k (128 blocks x 256 threads = 8 wave32s)
//  - full 128x128 f32 slice accumulator in LDS (64 KB) -> ds_add_f32 only,
//    zero global atomics (CDNA5 320 KB LDS: 78 KB/WG -> 4 WG/WGP resident)
//  - per-wave double-buffered LOR staging via global_load_async_to_lds_b128,
//    paced with s_wait_asynccnt (2 async instructions per 32-LOR chunk) --
//    no workgroup barriers in the main loop at all
//  - separable 3x3 Gaussian: 6 v_exp_f32 per (lor,slice) instead of 9

#define NV     128          // harness volume fixed at 128^3 (NVOX)
#define TPB    256
#define NWAVE  (TPB / 32)   // 8 wave32s
#define CHW    32           // LORs per wave-chunk (1 per lane)
#define ROWF   (7 * CHW)    // floats per staged chunk (6 coords + proj)

// 0.5 / kw^2 with kw^2 = 9/pi  ->  pi/18
#define CC 0.17453292519943295f

__device__ __forceinline__ void async_stage16(const float* src, float* lds_dst) {
#if defined(__HIP_DEVICE_COMPILE__)
  // Low 32 bits of a generic pointer to LDS = wave-relative LDS byte offset.
  unsigned lds_off = (unsigned)(unsigned long long)lds_dst;
  asm volatile("global_load_async_to_lds_b128 %0, %1, off"
               :: "v"(lds_off), "v"(src) : "memory");
#else
  (void)src; (void)lds_dst;
#endif
}
__device__ __forceinline__ void wait_async_le2() {
#if defined(__HIP_DEVICE_COMPILE__)
  asm volatile("s_wait_asynccnt 0x2" ::: "memory");
#endif
}
__device__ __forceinline__ void wait_async_le0() {
#if defined(__HIP_DEVICE_COMPILE__)
  asm volatile("s_wait_asynccnt 0x0" ::: "memory");
#endif
}
__device__ __forceinline__ void wait_ds0() {
#if defined(__HIP_DEVICE_COMPILE__)
  asm volatile("s_wait_dscnt 0x0" ::: "memory");
#endif
}

// Stage one 32-LOR chunk (7 rows x 32 floats = 56 x 16B units) with exactly
// two async-to-LDS instructions per wave: u = lane, and u = 32+lane (lane<24).
__device__ __forceinline__ void issue_unit(int u, int base, int nlor,
                                           const float* __restrict__ lors,
                                           const float* __restrict__ prj,
                                           float* stageBuf) {
  const int row = u >> 3;            // 8 x 16B units per 32-float row
  const int sub = (u & 7) << 2;      // float offset within row
  int col = base + sub;
  if (col > nlor - 4) col = nlor - 4;  // clamped tail, stays 16B-aligned
  const float* src = (row < 6) ? (lors + (size_t)row * (size_t)nlor + col)
                               : (prj + col);
  async_stage16(src, stageBuf + row * CHW + sub);
}
__device__ __forceinline__ void issue_chunk(int lane, int base, int nlor,
                                            const float* __restrict__ lors,
                                            const float* __restrict__ prj,
                                            float* stageBuf) {
  issue_unit(lane, base, nlor, lors, prj, stageBuf);          // asynccnt +1
  if (lane < 24)
    issue_unit(32 + lane, base, nlor, lors, prj, stageBuf);   // asynccnt +1
}

__global__ __launch_bounds__(TPB) void bp_slice(
    const float* __restrict__ lors,   // (6, nlor): p1x,p1y,p1z,p2x,p2y,p2z rows
    const float* __restrict__ prj,    // (nlor,)
    float* __restrict__ out,
    const float* __restrict__ center,
    const float* __restrict__ sizev,
    const int*   __restrict__ grid,
    int p0, int p1, int p2,           // axis permutation for geometry params
    int sk, int sy, int sz,           // output strides for (k, iy, iz)
    int nlor, int accumulate)
{
  __shared__ float acc[NV * NV];                 // 64 KB slice accumulator
  __shared__ float stage[NWAVE * 2 * ROWF];      // 14 KB: per-wave double buffer

  const int tid  = threadIdx.x;
  const int lane = tid & 31;
  const int wave = tid >> 5;
  const int k    = blockIdx.x;

  // Uniform geometry (permuted exactly like the reference).
  const float s0 = sizev[p0],  s1 = sizev[p1],  s2 = sizev[p2];
  const float c0 = center[p0], c1 = center[p1], c2 = center[p2];
  const int   n0 = grid[p0],   n1 = grid[p1],   n2 = grid[p2];
  const float v0 = s0 / (float)n0;
  const float v1 = s1 / (float)n1;
  const float v2 = s2 / (float)n2;
  const float lo0 = c0 - 0.5f * s0;
  const float lo1 = c1 - 0.5f * s1;
  const float lo2 = c2 - 0.5f * s2;
  const float xk  = lo0 + ((float)k + 0.5f) * v0;
  const float iv1 = 1.0f / v1;
  const float iv2 = 1.0f / v2;

  float* st[2] = { &stage[(wave * 2 + 0) * ROWF],
                   &stage[(wave * 2 + 1) * ROWF] };

  const int nch = (nlor + CHW - 1) / CHW;

  // Prologue: issue this wave's first chunk while the WG zeroes the slice.
  if (wave < nch) issue_chunk(lane, wave * CHW, nlor, lors, prj, st[0]);
  for (int i = tid; i < NV * NV; i += TPB) acc[i] = 0.0f;
  __syncthreads();   // acc zeroed (ds stores drained); stage not touched here

  int ibuf = 0;
  for (int c = wave; c < nch; c += NWAVE, ibuf ^= 1) {
    float* cur = st[ibuf];
    float* nxt = st[ibuf ^ 1];
    const int cn = c + NWAVE;

    if (cn < nch) {
      issue_chunk(lane, cn * CHW, nlor, lors, prj, nxt);
      wait_async_le2();   // in-order: all but the 2 newest done => cur resident
    } else {
      wait_async_le0();
    }

    // Pull this lane's LOR into registers, then retire the LDS reads so the
    // async engine may safely overwrite `cur` next iteration (ds pipe and
    // async-DMA LDS writes are not mutually ordered).
    const int j = c * CHW + lane;
    float p1x = 0.f, p1y = 0.f, p1z = 0.f, p2x = 1.f, p2y = 0.f, p2z = 0.f, pr = 0.f;
    if (j < nlor) {
      p1x = cur[0 * CHW + lane];
      p1y = cur[1 * CHW + lane];
      p1z = cur[2 * CHW + lane];
      p2x = cur[3 * CHW + lane];
      p2y = cur[4 * CHW + lane];
      p2z = cur[5 * CHW + lane];
      pr  = cur[6 * CHW + lane];
    }
    wait_ds0();

    if (j < nlor) {
      const float t  = (xk - p1x) / (p2x - p1x);
      const float yy = fmaf(t, p2y - p1y, p1y);
      const float zz = fmaf(t, p2z - p1z, p1z);
      const float fy = (yy - lo1) * iv1 - 0.5f;
      const float fz = (zz - lo2) * iv2 - 0.5f;
      const float iyf = floorf(fy);
      const float izf = floorf(fz);
      const int iy = (int)iyf - 1;          // leftmost of 3-tap footprint
      const int iz = (int)izf - 1;
      const float prt = (t >= 0.0f && t <= 1.0f) ? pr : 0.0f;

      float dy = fmaf(iyf - 0.5f, v1, lo1) - yy;   // center distance at o=0
      float dz = fmaf(izf - 0.5f, v2, lo2) - zz;
      float wy[3], wz[3];
      int iyc[3], izc[3];
      #pragma unroll
      for (int o = 0; o < 3; ++o) {
        const int a = iy + o, b = iz + o;
        wy[o] = (a >= 0 && a < n1) ? prt * __expf(-CC * dy * dy) : 0.0f;
        wz[o] = (b >= 0 && b < n2) ? __expf(-CC * dz * dz)       : 0.0f;
        iyc[o] = min(max(a, 0), n1 - 1);
        izc[o] = min(max(b, 0), n2 - 1);
        dy += v1; dz += v2;
      }
      // 3x3 outer-product splat into the LDS slice (ds_add_f32, no return).
      #pragma unroll
      for (int a = 0; a < 3; ++a) {
        #pragma unroll
        for (int b = 0; b < 3; ++b) {
          atomicAdd(&acc[iyc[a] * NV + izc[b]], wy[a] * wz[b]);
        }
      }
    }
  }

  __syncthreads();   // all waves' ds_adds retired

  // Write the slice out; workgroups own disjoint voxel sets within a kernel,
  // and the three direction kernels are ordered on the stream -> plain RMW.
  for (int i = tid; i < NV * NV; i += TPB) {
    const int iy = i >> 7;
    const int iz = i & (NV - 1);
    const int off = k * sk + iy * sy + iz * sz;
    if (accumulate) out[off] += acc[i];
    else            out[off]  = acc[i];
  }
}

extern "C" void kernel_launch(void* const* d_in, const int* in_sizes, int n_in,
                              void* d_out, int out_size, void* d_ws, size_t ws_size,
                              hipStream_t stream) {
  (void)n_in; (void)d_ws; (void)ws_size; (void)out_size;
  // setup_inputs() order:
  // 0 image (unused by reference output), 1 grid(i32 x3), 2 center(f32 x3),
  // 3 size(f32 x3), 4 xlors(6*n), 5 ylors, 6 zlors, 7 xproj(n), 8 yproj, 9 zproj
  const int*   grid   = (const int*)  d_in[1];
  const float* center = (const float*)d_in[2];
  const float* sizev  = (const float*)d_in[3];
  const float* xl     = (const float*)d_in[4];
  const float* yl     = (const float*)d_in[5];
  const float* zl     = (const float*)d_in[6];
  const float* xp     = (const float*)d_in[7];
  const float* yp     = (const float*)d_in[8];
  const float* zp     = (const float*)d_in[9];
  float* out = (float*)d_out;
  const int nlor = in_sizes[7];
  const int N = NV, NN = NV * NV;

  // z-direction: perm (0,1,2), out[k,iy,iz]; plain stores initialize d_out.
  hipLaunchKernelGGL(bp_slice, dim3(N), dim3(TPB), 0, stream,
                     zl, zp, out, center, sizev, grid, 0, 1, 2, NN, N, 1, nlor, 0);
  // y-direction: perm (1,0,2), out[iy,k,iz]; accumulate.
  hipLaunchKernelGGL(bp_slice, dim3(N), dim3(TPB), 0, stream,
                     yl, yp, out, center, sizev, grid, 1, 0, 2, N, NN, 1, nlor, 1);
  // x-direction: perm (2,0,1), out[iy,iz,k]; accumulate.
  hipLaunchKernelGGL(bp_slice, dim3(N), dim3(TPB), 0, stream,
                     xl, xp, out, center, sizev, grid, 2, 0, 1, 1, NN, N, nlor, 1);
}